// GraphEncoder_11141145166473
// MI455X (gfx1250) — compile-verified
//
#include <hip/hip_runtime.h>
#include <hip/hip_bf16.h>

#define N_NODES 30000
#define E_EDGES 300000
#define ETOT    330000      // E + N (self loops)
#define F_INP   16
#define F_EDGE  8
#define H_HEADS 4
#define C_CH    64
#define HID     256
#define G_GR    64
#define NEG_SLOPE 0.2f

typedef __attribute__((ext_vector_type(2))) float v2f;
typedef __attribute__((ext_vector_type(8))) float v8f;

// Guaranteed-hardware float atomic add (GLOBAL_ATOMIC_ADD_F32), no CAS fallback loop.
__device__ __forceinline__ void atomAddF(float* p, float v) {
    unsafeAtomicAdd(p, v);
}

// ---------------- utility fills ----------------
__global__ void fill_f32_kernel(float* __restrict__ p, float v, int n) {
    int i = blockIdx.x * blockDim.x + threadIdx.x;
    if (i < n) p[i] = v;
}
__global__ void fill_u32_kernel(unsigned* __restrict__ p, unsigned v, int n) {
    int i = blockIdx.x * blockDim.x + threadIdx.x;
    if (i < n) p[i] = v;
}

// ordered-uint float max atomics (monotone encoding) -> native integer atomicMax
__device__ __forceinline__ unsigned enc_ord(float f) {
    unsigned k = __float_as_uint(f);
    return (k & 0x80000000u) ? ~k : (k | 0x80000000u);
}
__device__ __forceinline__ float dec_ord(unsigned k) {
    k = (k & 0x80000000u) ? (k & 0x7FFFFFFFu) : ~k;
    return __uint_as_float(k);
}

// ---------------- self-loop attr prep ----------------
__global__ void deg_slsum_kernel(const int* __restrict__ edge_index,
                                 const float* __restrict__ edge_attr,
                                 float* __restrict__ deg, float* __restrict__ slsum) {
    int e = blockIdx.x * blockDim.x + threadIdx.x;
    if (e >= E_EDGES) return;
    int col = edge_index[E_EDGES + e];
    atomAddF(&deg[col], 1.0f);
    #pragma unroll
    for (int j = 0; j < F_EDGE; ++j)
        atomAddF(&slsum[col * F_EDGE + j], edge_attr[(size_t)e * F_EDGE + j]);
}
__global__ void sl_div_kernel(const float* __restrict__ slsum,
                              const float* __restrict__ deg,
                              float* __restrict__ slattr) {
    int i = blockIdx.x * blockDim.x + threadIdx.x;
    if (i >= N_NODES * F_EDGE) return;
    slattr[i] = slsum[i] / fmaxf(deg[i >> 3], 1.0f);
}

// ---------------- dual GEMM: xl = A@Wl+bl, xr = A@Wr+br via WMMA f32 16x16x4 --------
// block = 256 threads (8 waves). Block owns one 16-row M tile, all 256 cols of BOTH
// outputs: 32 16x16 tiles total, 4 per wave. A panel staged in LDS.
__global__ __launch_bounds__(256)
void gemm_dual_kernel(const float* __restrict__ A,
                      const float* __restrict__ Wl, const float* __restrict__ bl,
                      const float* __restrict__ Wr, const float* __restrict__ br,
                      float* __restrict__ xl, float* __restrict__ xr, int din) {
    extern __shared__ float ldsA[];            // 16 * din floats
    const int tid = threadIdx.x;
    const int rowbase = blockIdx.x * 16;
    const int total = 16 * din;
    for (int i = tid; i < total; i += 256) {
        int r = i / din, k = i - r * din;
        ldsA[i] = A[(size_t)(rowbase + r) * din + k];
    }
    __syncthreads();

    const int wave = tid >> 5;
    const int lane = tid & 31;
    const int m    = lane & 15;
    const int ksel = (lane >> 4) * 2;          // 0 or 2
    const int t0   = wave * 4;                 // flattened tile id: [0,16)=Wl, [16,32)=Wr

    v8f acc[4] = {};
    for (int k = 0; k < din; k += 4) {
        v2f a;
        a.x = ldsA[m * din + k + ksel];
        a.y = ldsA[m * din + k + ksel + 1];
        #pragma unroll
        for (int t = 0; t < 4; ++t) {
            int tile = t0 + t;
            const float* W = (tile < 16) ? Wl : Wr;
            int ncol = (tile & 15) * 16 + (lane & 15);
            v2f b;
            b.x = W[(size_t)(k + ksel) * HID + ncol];
            b.y = W[(size_t)(k + ksel + 1) * HID + ncol];
            acc[t] = __builtin_amdgcn_wmma_f32_16x16x4_f32(
                false, a, false, b, (short)0, acc[t], false, false);
        }
    }
    const int mb = (lane >> 4) * 8;
    #pragma unroll
    for (int t = 0; t < 4; ++t) {
        int tile = t0 + t;
        float* O = (tile < 16) ? xl : xr;
        const float* bias = (tile < 16) ? bl : br;
        int ncol = (tile & 15) * 16 + (lane & 15);
        float bv = bias[ncol];
        #pragma unroll
        for (int v = 0; v < 8; ++v)
            O[(size_t)(rowbase + mb + v) * HID + ncol] = acc[t][v] + bv;
    }
}

// ---------------- edge logits + segment max (wave per edge, We fused in LDS) --------
__global__ __launch_bounds__(256)
void edge_logits_kernel(const float* __restrict__ xl, const float* __restrict__ xr,
                        const int* __restrict__ edge_index,
                        const float* __restrict__ edge_attr,
                        const float* __restrict__ slattr,
                        const float* __restrict__ We, const float* __restrict__ attv,
                        float* __restrict__ logit, unsigned* __restrict__ smax) {
    __shared__ float sWe[F_EDGE * HID];
    __shared__ float sAtt[HID];
    const int tid = threadIdx.x;
    for (int i = tid; i < F_EDGE * HID; i += 256) sWe[i] = We[i];
    sAtt[tid] = attv[tid];
    __syncthreads();

    const int wave = tid >> 5, lane = tid & 31;
    const int e = blockIdx.x * 8 + wave;
    int row, col;
    const float* ea;
    if (e < E_EDGES) {
        row = edge_index[e];
        col = edge_index[E_EDGES + e];
        ea  = edge_attr + (size_t)e * F_EDGE;
    } else {
        row = col = e - E_EDGES;
        ea  = slattr + (size_t)row * F_EDGE;
    }
    float eav[F_EDGE];
    #pragma unroll
    for (int j = 0; j < F_EDGE; ++j) eav[j] = ea[j];

    const int c0 = lane * 8;
    // 16-byte gathers -> global_load_b128
    const float4* pl4 = (const float4*)(xl + (size_t)row * HID + c0);
    const float4* pr4 = (const float4*)(xr + (size_t)col * HID + c0);
    float4 l0 = pl4[0], l1 = pl4[1];
    float4 r0 = pr4[0], r1 = pr4[1];
    float lv[8] = {l0.x, l0.y, l0.z, l0.w, l1.x, l1.y, l1.z, l1.w};
    float rv[8] = {r0.x, r0.y, r0.z, r0.w, r1.x, r1.y, r1.z, r1.w};

    float part = 0.f;
    #pragma unroll
    for (int u = 0; u < 8; ++u) {
        float em = 0.f;
        #pragma unroll
        for (int j = 0; j < F_EDGE; ++j) em += eav[j] * sWe[j * HID + c0 + u];
        float mv = lv[u] + rv[u] + em;
        mv = (mv > 0.f) ? mv : NEG_SLOPE * mv;
        part += sAtt[c0 + u] * mv;
    }
    // 8 lanes per head (channels are head-contiguous): tree reduce in head group
    part += __shfl_xor(part, 1, 32);
    part += __shfl_xor(part, 2, 32);
    part += __shfl_xor(part, 4, 32);
    if ((lane & 7) == 0) {
        int head = lane >> 3;
        logit[(size_t)e * H_HEADS + head] = part;
        atomicMax(&smax[(size_t)col * H_HEADS + head], enc_ord(part));
    }
}

// ---------------- exp + segment sum ----------------
__global__ void edge_softmax_kernel(const float* __restrict__ logit,
                                    const unsigned* __restrict__ smax,
                                    const int* __restrict__ edge_index,
                                    float* __restrict__ exv, float* __restrict__ ssum) {
    int idx = blockIdx.x * blockDim.x + threadIdx.x;
    if (idx >= ETOT * H_HEADS) return;
    int e = idx >> 2, h = idx & 3;
    int col = (e < E_EDGES) ? edge_index[E_EDGES + e] : (e - E_EDGES);
    float mx = dec_ord(smax[col * H_HEADS + h]);
    float ex = __expf(logit[idx] - mx);
    exv[idx] = ex;
    atomAddF(&ssum[col * H_HEADS + h], ex);
}

// ---------------- weighted scatter aggregation (wave per edge) ----------------
__global__ __launch_bounds__(256)
void edge_aggregate_kernel(const float* __restrict__ xl, const float* __restrict__ exv,
                           const float* __restrict__ ssum,
                           const int* __restrict__ edge_index, float* __restrict__ acc) {
    const int tid = threadIdx.x;
    const int wave = tid >> 5, lane = tid & 31;
    const int e = blockIdx.x * 8 + wave;
    int row, col;
    if (e < E_EDGES) { row = edge_index[e]; col = edge_index[E_EDGES + e]; }
    else             { row = col = e - E_EDGES; }
    const int head = lane >> 3;
    float alpha = exv[(size_t)e * H_HEADS + head] / ssum[(size_t)col * H_HEADS + head];
    const int c0 = lane * 8;
    const float4* pl4 = (const float4*)(xl + (size_t)row * HID + c0);
    float4 l0 = pl4[0], l1 = pl4[1];
    float lv[8] = {l0.x, l0.y, l0.z, l0.w, l1.x, l1.y, l1.z, l1.w};
    float* po = acc + (size_t)col * HID + c0;
    #pragma unroll
    for (int u = 0; u < 8; ++u) atomAddF(&po[u], lv[u] * alpha);
}

// ---------------- bias + optional relu (in place) ----------------
__global__ void bias_act_kernel(float* __restrict__ acc, const float* __restrict__ bo,
                                int do_relu) {
    int idx = blockIdx.x * blockDim.x + threadIdx.x;
    if (idx >= N_NODES * HID) return;
    float v = acc[idx] + bo[idx & (HID - 1)];
    acc[idx] = do_relu ? fmaxf(v, 0.f) : v;
}

// ---------------- global mean pool ----------------
__global__ __launch_bounds__(256)
void pool_accum_kernel(const float* __restrict__ h, const int* __restrict__ batch,
                       float* __restrict__ out, float* __restrict__ cnt) {
    const int tid = threadIdx.x;
    const int wave = tid >> 5, lane = tid & 31;
    const int n = blockIdx.x * 8 + wave;
    if (n >= N_NODES) return;
    int g = batch[n];
    if (lane == 0) atomAddF(&cnt[g], 1.0f);
    const int c0 = lane * 8;
    const float4* ph4 = (const float4*)(h + (size_t)n * HID + c0);
    float4 h0 = ph4[0], h1 = ph4[1];
    float hv[8] = {h0.x, h0.y, h0.z, h0.w, h1.x, h1.y, h1.z, h1.w};
    float* po = out + (size_t)g * HID + c0;
    #pragma unroll
    for (int u = 0; u < 8; ++u) atomAddF(&po[u], hv[u]);
}
__global__ void pool_finalize_kernel(float* __restrict__ out, const float* __restrict__ cnt) {
    int idx = blockIdx.x * blockDim.x + threadIdx.x;
    if (idx >= G_GR * HID) return;
    out[idx] /= fmaxf(cnt[idx >> 8], 1.0f);
}

// ---------------- host orchestration ----------------
extern "C" void kernel_launch(void* const* d_in, const int* in_sizes, int n_in,
                              void* d_out, int out_size, void* d_ws, size_t ws_size,
                              hipStream_t stream) {
    const float* x          = (const float*)d_in[0];
    const int*   edge_index = (const int*)  d_in[1];
    const float* edge_attr  = (const float*)d_in[2];
    const int*   batch      = (const int*)  d_in[3];
    const float *Wl[3], *bl[3], *Wr[3], *br[3], *We[3], *att[3], *bo[3];
    for (int l = 0; l < 3; ++l) {
        int b = 4 + l * 7;
        Wl[l]  = (const float*)d_in[b + 0];
        bl[l]  = (const float*)d_in[b + 1];
        Wr[l]  = (const float*)d_in[b + 2];
        br[l]  = (const float*)d_in[b + 3];
        We[l]  = (const float*)d_in[b + 4];
        att[l] = (const float*)d_in[b + 5];
        bo[l]  = (const float*)d_in[b + 6];
    }

    const size_t NH = (size_t)N_NODES * HID;
    float* ws = (float*)d_ws;
    size_t off = 0;
    auto carve = [&](size_t n) { float* p = ws + off; off += n; return p; };
    float*    xl     = carve(NH);
    float*    xr     = carve(NH);
    float*    hP     = carve(NH);
    float*    hQ     = carve(NH);
    float*    logit  = carve((size_t)ETOT * H_HEADS);
    float*    exv    = carve((size_t)ETOT * H_HEADS);
    unsigned* smax   = (unsigned*)carve((size_t)N_NODES * H_HEADS);
    float*    ssum   = carve((size_t)N_NODES * H_HEADS);
    float*    deg    = carve(N_NODES);
    float*    slsum  = carve((size_t)N_NODES * F_EDGE);
    float*    slattr = carve((size_t)N_NODES * F_EDGE);
    float*    cnt    = carve(G_GR);
    float*    out    = (float*)d_out;

    const int T = 256;
    auto blocks = [](long n) { return (int)((n + 255) / 256); };

    // self-loop edge attrs (mean of incoming)
    fill_f32_kernel<<<blocks(N_NODES), T, 0, stream>>>(deg, 0.f, N_NODES);
    fill_f32_kernel<<<blocks((long)N_NODES * F_EDGE), T, 0, stream>>>(slsum, 0.f, N_NODES * F_EDGE);
    deg_slsum_kernel<<<blocks(E_EDGES), T, 0, stream>>>(edge_index, edge_attr, deg, slsum);
    sl_div_kernel<<<blocks((long)N_NODES * F_EDGE), T, 0, stream>>>(slsum, deg, slattr);

    const float* hin = x;
    int din = F_INP;
    float* cur = hQ; float* nxt = hP;
    const unsigned NEG_INF_ENC = 0x007FFFFFu;   // enc_ord(-inf)

    for (int l = 0; l < 3; ++l) {
        gemm_dual_kernel<<<N_NODES / 16, T, 16 * din * sizeof(float), stream>>>(
            hin, Wl[l], bl[l], Wr[l], br[l], xl, xr, din);
        fill_u32_kernel<<<blocks(N_NODES * H_HEADS), T, 0, stream>>>(smax, NEG_INF_ENC, N_NODES * H_HEADS);
        fill_f32_kernel<<<blocks(N_NODES * H_HEADS), T, 0, stream>>>(ssum, 0.f, N_NODES * H_HEADS);
        fill_f32_kernel<<<blocks((long)NH), T, 0, stream>>>(cur, 0.f, (int)NH);
        edge_logits_kernel<<<ETOT / 8, T, 0, stream>>>(
            xl, xr, edge_index, edge_attr, slattr, We[l], att[l], logit, smax);
        edge_softmax_kernel<<<blocks((long)ETOT * H_HEADS), T, 0, stream>>>(
            logit, smax, edge_index, exv, ssum);
        edge_aggregate_kernel<<<ETOT / 8, T, 0, stream>>>(xl, exv, ssum, edge_index, cur);
        bias_act_kernel<<<blocks((long)NH), T, 0, stream>>>(cur, bo[l], l < 2 ? 1 : 0);
        hin = cur; din = HID;
        float* t = cur; cur = nxt; nxt = t;
    }

    fill_f32_kernel<<<blocks(G_GR * HID), T, 0, stream>>>(out, 0.f, G_GR * HID);
    fill_f32_kernel<<<1, G_GR, 0, stream>>>(cnt, 0.f, G_GR);
    pool_accum_kernel<<<N_NODES / 8, T, 0, stream>>>(hin, batch, out, cnt);
    pool_finalize_kernel<<<blocks(G_GR * HID), T, 0, stream>>>(out, cnt);
}